// Link_Prediction_GNN_34411277976464
// MI455X (gfx1250) — compile-verified
//
#include <hip/hip_runtime.h>

// ---------------------------------------------------------------------------
// GraphSAGE 2-layer forward for MI455X (gfx1250, wave32, WMMA)
//   layer: out = lin_l(mean_agg(x)) + lin_r(x)   (+bias on lin_l, ReLU on L1)
// GEMMs use V_WMMA_F32_16X16X4_F32 (full fp32 precision).
// ---------------------------------------------------------------------------

typedef __attribute__((ext_vector_type(2))) float v2f;
typedef __attribute__((ext_vector_type(8))) float v8f;

#define D 128

// ---------------- zeroing ----------------
__global__ void fill_zero_kernel(float* __restrict__ p, int n) {
    int i = blockIdx.x * blockDim.x + threadIdx.x;
    if (i < n) p[i] = 0.0f;
}

// ---------------- degree count ----------------
__global__ void count_deg_kernel(const int* __restrict__ dst,
                                 float* __restrict__ cnt, int E) {
    int e = blockIdx.x * blockDim.x + threadIdx.x;
    if (e < E) atomicAdd(&cnt[dst[e]], 1.0f);
}

// ---------------- edge gather + scatter-add ----------------
// 32 threads per edge, 4 floats per thread (covers D=128).
__global__ void scatter_add_kernel(const float* __restrict__ xin,
                                   const int* __restrict__ src,
                                   const int* __restrict__ dst,
                                   float* __restrict__ agg, int E) {
    int tid = blockIdx.x * blockDim.x + threadIdx.x;
    int e = tid >> 5;
    if (e >= E) return;
    int c = (tid & 31) * 4;
    int s = src[e];
    int d = dst[e];
    const float4 v = *(const float4*)(xin + (size_t)s * D + c);
    float* p = agg + (size_t)d * D + c;
    atomicAdd(p + 0, v.x);
    atomicAdd(p + 1, v.y);
    atomicAdd(p + 2, v.z);
    atomicAdd(p + 3, v.w);
}

// ---------------- fused SAGE GEMM ----------------
// One wave handles a 16-row block:
//   out[16, 128] = (agg/max(cnt,1)) @ Wl^T + bl + xin @ Wr^T   (ReLU optional)
// A-frag (16x4 f32): lane holds rows (lane&15), K = k0 + 2*(lane>>4) .. +1
// B-frag (4x16 f32): lane holds col  (lane&15), K = k0 + 2*(lane>>4) .. +1
//   -> B = W^T tile means loading W row-major: W[col][k] -- identical pattern.
// C/D (16x16 f32, 8 VGPRs): vgpr r -> row r + 8*(lane>>4), col (lane&15).
template <bool RELU>
__global__ __launch_bounds__(64) void sage_gemm_kernel(
    const float* __restrict__ agg, const float* __restrict__ cnt,
    const float* __restrict__ xin, const float* __restrict__ Wl,
    const float* __restrict__ bl, const float* __restrict__ Wr,
    float* __restrict__ out, int nrows) {
    const int lane = threadIdx.x & 31;
    const int wave = threadIdx.x >> 5;
    const int row0 = (blockIdx.x * 2 + wave) * 16;
    if (row0 >= nrows) return;  // uniform per wave: EXEC stays all-1s

    const int arow  = lane & 15;        // A row / B col within tile
    const int akoff = (lane >> 4) * 2;  // k sub-offset: 0 or 2

    const float c = cnt[row0 + arow];
    const float scale = 1.0f / fmaxf(c, 1.0f);

    const float* aM = agg + (size_t)(row0 + arow) * D + akoff;
    const float* aX = xin + (size_t)(row0 + arow) * D + akoff;
    const float* wL = Wl + (size_t)arow * D + akoff;  // + t*16*D per col tile
    const float* wR = Wr + (size_t)arow * D + akoff;

    v8f acc[8];
#pragma unroll
    for (int t = 0; t < 8; ++t) {
        v8f z = {};
        acc[t] = z;
    }

    for (int k0 = 0; k0 < D; k0 += 4) {
        v2f am = *(const v2f*)(aM + k0);
        am = am * scale;                     // fold mean division into A-frag
        const v2f ax = *(const v2f*)(aX + k0);
#pragma unroll
        for (int t = 0; t < 8; ++t) {
            const v2f bL = *(const v2f*)(wL + (size_t)t * 16 * D + k0);
            const v2f bR = *(const v2f*)(wR + (size_t)t * 16 * D + k0);
            acc[t] = __builtin_amdgcn_wmma_f32_16x16x4_f32(
                false, am, false, bL, (short)0, acc[t], false, false);
            acc[t] = __builtin_amdgcn_wmma_f32_16x16x4_f32(
                false, ax, false, bR, (short)0, acc[t], false, false);
        }
    }

    // epilogue: bias (+ReLU) and store
    const int col   = lane & 15;
    const int rhalf = (lane >> 4) * 8;
#pragma unroll
    for (int t = 0; t < 8; ++t) {
        const int c0 = t * 16 + col;
        const float bias = bl[c0];
#pragma unroll
        for (int r = 0; r < 8; ++r) {
            float v = acc[t][r] + bias;
            if (RELU) v = fmaxf(v, 0.0f);
            out[(size_t)(row0 + r + rhalf) * D + c0] = v;
        }
    }
}

// ---------------------------------------------------------------------------
extern "C" void kernel_launch(void* const* d_in, const int* in_sizes, int n_in,
                              void* d_out, int out_size, void* d_ws,
                              size_t ws_size, hipStream_t stream) {
    const float* x   = (const float*)d_in[0];
    const int*   ei  = (const int*)d_in[1];
    const float* Wl1 = (const float*)d_in[2];
    const float* bl1 = (const float*)d_in[3];
    const float* Wr1 = (const float*)d_in[4];
    const float* Wl2 = (const float*)d_in[5];
    const float* bl2 = (const float*)d_in[6];
    const float* Wr2 = (const float*)d_in[7];
    float* out = (float*)d_out;

    const int N = in_sizes[0] / D;  // 100000
    const int E = in_sizes[1] / 2;  // 1600000
    const int* src = ei;
    const int* dst = ei + E;

    // workspace layout: cnt [N] | agg [N*D] | h [N*D]
    float* cnt = (float*)d_ws;
    float* agg = cnt + (size_t)((N + 63) & ~63);
    float* h   = agg + (size_t)N * D;

    const int ND = N * D;
    const dim3 blk256(256);
    const int zero_grid_nd  = (ND + 255) / 256;
    const int zero_grid_n   = (N + 255) / 256;
    const int count_grid    = (E + 255) / 256;
    const long sthreads     = (long)E * 32;
    const int scatter_grid  = (int)((sthreads + 255) / 256);
    const int nblk16        = (N + 15) / 16;      // 6250 (exact, no tail)
    const int gemm_grid     = (nblk16 + 1) / 2;   // 2 waves / block

    // ---- degree counts (shared by both layers) ----
    fill_zero_kernel<<<zero_grid_n, blk256, 0, stream>>>(cnt, N);
    count_deg_kernel<<<count_grid, blk256, 0, stream>>>(dst, cnt, E);

    // ---- layer 1: h = relu(mean_agg(x) @ Wl1^T + bl1 + x @ Wr1^T) ----
    fill_zero_kernel<<<zero_grid_nd, blk256, 0, stream>>>(agg, ND);
    scatter_add_kernel<<<scatter_grid, blk256, 0, stream>>>(x, src, dst, agg, E);
    sage_gemm_kernel<true><<<gemm_grid, dim3(64), 0, stream>>>(
        agg, cnt, x, Wl1, bl1, Wr1, h, N);

    // ---- layer 2: out = mean_agg(h) @ Wl2^T + bl2 + h @ Wr2^T ----
    fill_zero_kernel<<<zero_grid_nd, blk256, 0, stream>>>(agg, ND);
    scatter_add_kernel<<<scatter_grid, blk256, 0, stream>>>(h, src, dst, agg, E);
    sage_gemm_kernel<false><<<gemm_grid, dim3(64), 0, stream>>>(
        agg, cnt, h, Wl2, bl2, Wr2, out, N);
}